// CaNetConv_12970801234191
// MI455X (gfx1250) — compile-verified
//
#include <hip/hip_runtime.h>
#include <cstdint>
#include <cstddef>

#define N_NODES 50000
#define N_EDGES 800000
#define DIM     128
#define NENV    4
#define M_TOT   (N_EDGES + N_NODES)   // 850000 edges incl. self loops
#define EPSV    1e-8f

typedef __attribute__((ext_vector_type(2))) float v2f;
typedef __attribute__((ext_vector_type(8))) float v8f;

// order-preserving float<->int key (works for negative values; self-inverse)
__device__ __forceinline__ int f2key(float f) {
    int b = __float_as_int(f);
    return (b >= 0) ? b : (int)(0x80000000u - (unsigned)b);
}
__device__ __forceinline__ float key2f(int k) {
    int b = (k >= 0) ? k : (int)(0x80000000u - (unsigned)k);
    return __int_as_float(b);
}

// ---------------------------------------------------------------- init
__global__ void k_init_out(const float* __restrict__ x, float* __restrict__ out) {
    int i = blockIdx.x * blockDim.x + threadIdx.x;
    if (i < N_NODES * DIM) out[i] = x[i];
}

__global__ void k_init_stats(float* __restrict__ s_src, float* __restrict__ s_dst,
                             float* __restrict__ denom, int* __restrict__ gmax) {
    int i = blockIdx.x * blockDim.x + threadIdx.x;
    if (i < NENV * N_NODES) { s_src[i] = 0.f; s_dst[i] = 0.f; denom[i] = 0.f; }
    if (i < NENV) gmax[i] = (int)0x80000000;  // encoded -inf sentinel
}

// ---------------------------------------------------------------- GEMM: h = x @ W[e]  (fused s_src/s_dst)
// grid = (N/16, NENV), block = 256 (8 wave32s). Each wave owns a 16x16 tile of
// the 16-row x 128-col output strip; K=128 swept in steps of 4 with
// V_WMMA_F32_16X16X4_F32 (full f32 precision).
#define LXS 132   // padded LDS row stride: (132 mod 64)=4 => rows hit distinct banks

__global__ __launch_bounds__(256) void k_gemm_h(
    const float* __restrict__ x, const float* __restrict__ W,
    const float* __restrict__ a, float* __restrict__ h,
    float* __restrict__ s_src, float* __restrict__ s_dst) {
    __shared__ float lx[16 * LXS];

    const int e   = blockIdx.y;
    const int n0  = blockIdx.x * 16;
    const int tid = threadIdx.x;

    // cooperative load of the 16x128 x-tile
    for (int i = tid; i < 16 * DIM; i += 256) {
        int r = i >> 7, c = i & 127;
        lx[r * LXS + c] = (n0 + r < N_NODES) ? x[(size_t)(n0 + r) * DIM + c] : 0.f;
    }
    __syncthreads();

    const int wave = tid >> 5;
    const int lane = tid & 31;
    const int l16  = lane & 15;
    const int half = lane >> 4;        // 0: K-pair {k,k+1} / rows M..M+7 ; 1: {k+2,k+3} / M+8..
    const int f0   = wave * 16;
    const float* We = W + (size_t)e * DIM * DIM;

    v8f acc = {};
    #pragma unroll 4
    for (int kk = 0; kk < DIM; kk += 4) {
        const int kb = kk + 2 * half;
        v2f A, B;
        // A 16x4 f32: lanes 0-15 rows M=0..15 K={kk,kk+1}; lanes 16-31 K={kk+2,kk+3}
        A.x = lx[l16 * LXS + kb];
        A.y = lx[l16 * LXS + kb + 1];
        // B 4x16 f32: row K striped across 16 lanes, same half split as A
        B.x = We[(size_t)kb * DIM + f0 + l16];
        B.y = We[(size_t)(kb + 1) * DIM + f0 + l16];
        acc = __builtin_amdgcn_wmma_f32_16x16x4_f32(
            /*neg_a=*/false, A, /*neg_b=*/false, B,
            /*c_mod=*/(short)0, acc, /*reuse_a=*/false, /*reuse_b=*/false);
    }

    // attention vectors for this wave's 16 features
    const float a1v = a[e * (2 * DIM) + f0 + l16];
    const float a2v = a[e * (2 * DIM) + DIM + f0 + l16];

    // C layout: VGPR j holds row M=j (lanes 0-15) and M=j+8 (lanes 16-31), N=lane&15
    #pragma unroll
    for (int j = 0; j < 8; ++j) {
        const int row = j + 8 * half;
        const int n   = n0 + row;
        float c = acc[j];
        if (n < N_NODES)
            h[((size_t)e * N_NODES + n) * DIM + f0 + l16] = c;
        // fused partial dot products for s_src / s_dst (reduce across the 16-lane group)
        float p1 = c * a1v, p2 = c * a2v;
        #pragma unroll
        for (int off = 8; off > 0; off >>= 1) {
            p1 += __shfl_xor(p1, off, 16);
            p2 += __shfl_xor(p2, off, 16);
        }
        if (l16 == 0 && n < N_NODES) {
            atomicAdd(&s_src[e * N_NODES + n], p1);
            atomicAdd(&s_dst[e * N_NODES + n], p2);
        }
    }
}

// ---------------------------------------------------------------- per-env global max over edges
__global__ __launch_bounds__(256) void k_edge_max(
    const int* __restrict__ rows, const int* __restrict__ cols,
    const float* __restrict__ s_src, const float* __restrict__ s_dst,
    int* __restrict__ gmax) {
    int m = blockIdx.x * blockDim.x + threadIdx.x;
    float lm[NENV];
    #pragma unroll
    for (int e = 0; e < NENV; ++e) lm[e] = -3.0e38f;
    if (m < M_TOT) {
        int s, d;
        if (m < N_EDGES) { s = rows[m]; d = cols[m]; } else { s = d = m - N_EDGES; }
        #pragma unroll
        for (int e = 0; e < NENV; ++e) {
            float l = s_src[e * N_NODES + s] + s_dst[e * N_NODES + d];
            lm[e] = (l > 0.f) ? l : 0.01f * l;   // leaky_relu(0.01)
        }
    }
    #pragma unroll
    for (int e = 0; e < NENV; ++e) {
        float v = lm[e];
        #pragma unroll
        for (int off = 16; off > 0; off >>= 1) {
            float o = __shfl_xor(v, off, 32);
            v = (o > v) ? o : v;
        }
        if ((threadIdx.x & 31) == 0) atomicMax(&gmax[e], f2key(v));
    }
}

// ---------------------------------------------------------------- exp + denominator (segment sum over src)
__global__ __launch_bounds__(256) void k_edge_exp(
    const int* __restrict__ rows, const int* __restrict__ cols,
    const float* __restrict__ s_src, const float* __restrict__ s_dst,
    const int* __restrict__ gmax, float4* __restrict__ ek4,
    float* __restrict__ denom) {
    int m = blockIdx.x * blockDim.x + threadIdx.x;
    if (m >= M_TOT) return;
    int s, d;
    if (m < N_EDGES) { s = rows[m]; d = cols[m]; } else { s = d = m - N_EDGES; }
    float ev[NENV];
    #pragma unroll
    for (int e = 0; e < NENV; ++e) {
        float mx = key2f(gmax[e]);
        float l  = s_src[e * N_NODES + s] + s_dst[e * N_NODES + d];
        l = (l > 0.f) ? l : 0.01f * l;
        float ee = __expf(l - mx);
        ev[e] = ee;
        atomicAdd(&denom[e * N_NODES + s], ee);
    }
    ek4[m] = make_float4(ev[0], ev[1], ev[2], ev[3]);
}

// ---------------------------------------------------------------- scatter: out[src] += sum_e coef_e * h[e,dst,:]
// one wave32 per edge: 32 lanes x float4 = 128 features; h is L2-resident (102 MB < 192 MB L2)
__global__ __launch_bounds__(256) void k_scatter(
    const int* __restrict__ rows, const int* __restrict__ cols,
    const float* __restrict__ env_w, const float4* __restrict__ ek4,
    const float* __restrict__ denom, const float* __restrict__ h,
    float* __restrict__ out) {
    long long gt = (long long)blockIdx.x * blockDim.x + threadIdx.x;
    long long gw = gt >> 5;
    int lane = threadIdx.x & 31;
    if (gw >= M_TOT) return;
    int m = (int)gw;
    int s, d;
    if (m < N_EDGES) { s = rows[m]; d = cols[m]; } else { s = d = m - N_EDGES; }

    float4 ek = ek4[m];
    const float ekp[NENV] = { ek.x, ek.y, ek.z, ek.w };
    float coef[NENV];
    #pragma unroll
    for (int e = 0; e < NENV; ++e)
        coef[e] = env_w[(size_t)s * NENV + e] * ekp[e] / (denom[e * N_NODES + s] + EPSV);

    const int f0 = lane * 4;
    float a0 = 0.f, a1 = 0.f, a2 = 0.f, a3 = 0.f;
    #pragma unroll
    for (int e = 0; e < NENV; ++e) {
        const float4 hv = *(const float4*)&h[((size_t)e * N_NODES + d) * DIM + f0];
        a0 += coef[e] * hv.x; a1 += coef[e] * hv.y;
        a2 += coef[e] * hv.z; a3 += coef[e] * hv.w;
    }
    float* op = &out[(size_t)s * DIM + f0];
    atomicAdd(op + 0, a0); atomicAdd(op + 1, a1);
    atomicAdd(op + 2, a2); atomicAdd(op + 3, a3);
}

// ---------------------------------------------------------------- launch
extern "C" void kernel_launch(void* const* d_in, const int* in_sizes, int n_in,
                              void* d_out, int out_size, void* d_ws, size_t ws_size,
                              hipStream_t stream) {
    const float* x    = (const float*)d_in[0];
    const int*   adj  = (const int*)d_in[1];
    const float* envw = (const float*)d_in[2];
    const float* W    = (const float*)d_in[3];
    const float* a    = (const float*)d_in[4];
    float*       out  = (float*)d_out;

    const int* rows = adj;              // adj[0]
    const int* cols = adj + N_EDGES;    // adj[1]

    // workspace layout (floats)
    float* ws     = (float*)d_ws;
    float* h      = ws;                                            // NENV*N*D = 25.6M floats
    float* s_src  = h     + (size_t)NENV * N_NODES * DIM;          // NENV*N
    float* s_dst  = s_src + (size_t)NENV * N_NODES;
    float* denom  = s_dst + (size_t)NENV * N_NODES;
    int*   gmax   = (int*)(denom + (size_t)NENV * N_NODES);        // 4 ints, padded to 16 floats
    float4* ek4   = (float4*)(denom + (size_t)NENV * N_NODES + 16);// M_TOT float4

    // 1) init
    k_init_out  <<<(N_NODES * DIM + 255) / 256, 256, 0, stream>>>(x, out);
    k_init_stats<<<(NENV * N_NODES + 255) / 256, 256, 0, stream>>>(s_src, s_dst, denom, gmax);

    // 2) h = x @ W[e]  (WMMA f32) + fused s_src/s_dst
    dim3 ggemm(N_NODES / 16, NENV);   // 50000 % 16 == 0
    k_gemm_h<<<ggemm, 256, 0, stream>>>(x, W, a, h, s_src, s_dst);

    // 3) per-env max over all edges
    k_edge_max<<<(M_TOT + 255) / 256, 256, 0, stream>>>(rows, cols, s_src, s_dst, gmax);

    // 4) exp + denominator
    k_edge_exp<<<(M_TOT + 255) / 256, 256, 0, stream>>>(rows, cols, s_src, s_dst, gmax, ek4, denom);

    // 5) normalized weighted scatter into out (wave per edge)
    long long waves = (long long)M_TOT;
    int blocks = (int)((waves * 32 + 255) / 256);
    k_scatter<<<blocks, 256, 0, stream>>>(rows, cols, envw, ek4, denom, h, out);
}